// NavierStokesPINNLoss_53283364274442
// MI455X (gfx1250) — compile-verified
//
#include <hip/hip_runtime.h>
#include <hip/hip_bf16.h>

// NavierStokes PINN loss, fused single-pass stencil + hierarchical reduction.
// B=2, C=4 (u,v,w,p), D=H=W=128. x is the fastest axis.
#define NX 128
#define CH_STRIDE 2097152LL           // 128^3
#define NPTS 4194304LL                // 2 * 128^3
#define OFF_X 1
#define OFF_Y 128
#define OFF_Z 16384

#define INV2D  50.0f                  // 1/(2*0.01)
#define INVD2  10000.0f               // 1/0.01^2
#define INVRE  0.01f                  // 1/RE
#define C_INT  (1.0f / 4000752.0f)    // 1/(2*126^3): cont + mom means
#define C_NS   (1.0f / 4194304.0f)    // 1/(2*128^3): noslip mean
#define C_PL   (1.0f / 32768.0f)      // 1/(2*128^2): inlet/outlet plane means

#define S1_BLOCKS 2048
#define S1_THREADS 256

typedef __attribute__((ext_vector_type(2))) float v2f;
typedef __attribute__((ext_vector_type(8))) float v8f;

__global__ __launch_bounds__(S1_THREADS)
void pinn_stage1(const float* __restrict__ pred,
                 const float* __restrict__ sdf,
                 float* __restrict__ partials) {
    const int tid = threadIdx.x;
    const long long stride = (long long)gridDim.x * blockDim.x;
    float acc = 0.0f;

    for (long long i = (long long)blockIdx.x * blockDim.x + tid; i < NPTS; i += stride) {
        const int x = (int)(i & 127);
        const int y = (int)((i >> 7) & 127);
        const int z = (int)((i >> 14) & 127);
        const int b = (int)(i >> 21);

        // sdf flat index == i (single channel). pred u-channel index:
        const float* u = pred + (i + 3LL * CH_STRIDE * b);
        const float* v = u + CH_STRIDE;
        const float* w = v + CH_STRIDE;
        const float* p = w + CH_STRIDE;

        const float sd = sdf[i];
        const float u0 = u[0], v0 = v[0], w0 = w[0];

        // no-slip over full volume
        if (sd <= 0.0f)
            acc += (u0 * u0 + v0 * v0 + w0 * w0) * C_NS;

        // inlet plane x == 0
        if (x == 0) {
            const float du = u0 - 1.0f;
            acc += (du * du + v0 * v0 + w0 * w0) * C_PL;
        }
        // outlet plane x == 127 (difference with x == 126)
        if (x == NX - 1) {
            const float du = u0 - u[-OFF_X];
            const float dv = v0 - v[-OFF_X];
            const float dw = w0 - w[-OFF_X];
            acc += (du * du + dv * dv + dw * dw) * C_PL;
        }

        // continuity + momentum on fluid interior
        const bool interior = (x >= 1) & (x <= NX - 2) &
                              (y >= 1) & (y <= NX - 2) &
                              (z >= 1) & (z <= NX - 2);
        if (interior && sd > 0.0f) {
            const float uxm = u[-OFF_X], uxp = u[OFF_X];
            const float uym = u[-OFF_Y], uyp = u[OFF_Y];
            const float uzm = u[-OFF_Z], uzp = u[OFF_Z];
            const float vxm = v[-OFF_X], vxp = v[OFF_X];
            const float vym = v[-OFF_Y], vyp = v[OFF_Y];
            const float vzm = v[-OFF_Z], vzp = v[OFF_Z];
            const float wxm = w[-OFF_X], wxp = w[OFF_X];
            const float wym = w[-OFF_Y], wyp = w[OFF_Y];
            const float wzm = w[-OFF_Z], wzp = w[OFF_Z];
            const float pxm = p[-OFF_X], pxp = p[OFF_X];
            const float pym = p[-OFF_Y], pyp = p[OFF_Y];
            const float pzm = p[-OFF_Z], pzp = p[OFF_Z];

            const float dudx = (uxp - uxm) * INV2D;
            const float dudy = (uyp - uym) * INV2D;
            const float dudz = (uzp - uzm) * INV2D;
            const float dvdx = (vxp - vxm) * INV2D;
            const float dvdy = (vyp - vym) * INV2D;
            const float dvdz = (vzp - vzm) * INV2D;
            const float dwdx = (wxp - wxm) * INV2D;
            const float dwdy = (wyp - wym) * INV2D;
            const float dwdz = (wzp - wzm) * INV2D;
            const float dpdx = (pxp - pxm) * INV2D;
            const float dpdy = (pyp - pym) * INV2D;
            const float dpdz = (pzp - pzm) * INV2D;

            const float lapu = ((uxp - 2.0f * u0 + uxm) +
                                (uyp - 2.0f * u0 + uym) +
                                (uzp - 2.0f * u0 + uzm)) * INVD2;
            const float lapv = ((vxp - 2.0f * v0 + vxm) +
                                (vyp - 2.0f * v0 + vym) +
                                (vzp - 2.0f * v0 + vzm)) * INVD2;
            const float lapw = ((wxp - 2.0f * w0 + wxm) +
                                (wyp - 2.0f * w0 + wym) +
                                (wzp - 2.0f * w0 + wzm)) * INVD2;

            const float div = dudx + dvdy + dwdz;
            const float rx = u0 * dudx + v0 * dudy + w0 * dudz + dpdx - INVRE * lapu;
            const float ry = u0 * dvdx + v0 * dvdy + w0 * dvdz + dpdy - INVRE * lapv;
            const float rz = u0 * dwdx + v0 * dwdy + w0 * dwdz + dpdz - INVRE * lapw;

            acc += (div * div + rx * rx + ry * ry + rz * rz) * C_INT;
        }
    }

    // wave32 reduction
    #pragma unroll
    for (int off = 16; off > 0; off >>= 1)
        acc += __shfl_down(acc, off, 32);

    __shared__ float sm[S1_THREADS / 32];
    const int wave = tid >> 5;
    const int lane = tid & 31;
    if (lane == 0) sm[wave] = acc;
    __syncthreads();
    if (wave == 0) {
        float s = (lane < (S1_THREADS / 32)) ? sm[lane] : 0.0f;
        #pragma unroll
        for (int off = 4; off > 0; off >>= 1)
            s += __shfl_down(s, off, 32);
        if (lane == 0) partials[blockIdx.x] = s;
    }
}

// Stage 2: one wave32. 64 fp32 accumulator slots arranged as the 16x4 A-matrix
// (ISA layout: lanes 0-15 hold K=0 (a.x) and K=1 (a.y); lanes 16-31 hold
// K=2 / K=3). B = all ones => D[m,n] = sum_k A[m,k]. Column N=0 of D lives in
// lane 0 (M=0..7 across the 8 VGPRs) and lane 16 (M=8..15).
__global__ __launch_bounds__(32)
void pinn_stage2(const float* __restrict__ partials, float* __restrict__ out, int np) {
    const int lane = threadIdx.x;

#if defined(__gfx1250__) && __has_builtin(__builtin_amdgcn_wmma_f32_16x16x4_f32)
    v2f a;
    a.x = 0.0f;
    a.y = 0.0f;
    for (int i = lane * 2; i < np; i += 64)       a.x += partials[i];
    for (int i = lane * 2 + 1; i < np; i += 64)   a.y += partials[i];

    v2f bones;
    bones.x = 1.0f;
    bones.y = 1.0f;
    v8f c = {};
    // (neg_a, A, neg_b, B, c_mod, C, reuse_a, reuse_b)
    c = __builtin_amdgcn_wmma_f32_16x16x4_f32(
            false, a, false, bones, (short)0, c, false, false);

    float t = c[0] + c[1] + c[2] + c[3] + c[4] + c[5] + c[6] + c[7];
    const float t_hi = __shfl(t, 16, 32);   // M=8..15 of column 0
    const float t_lo = __shfl(t, 0, 32);    // M=0..7  of column 0
    if (lane == 0) out[0] = t_lo + t_hi;
#else
    // Fallback: plain wave32 tree reduction (also used by the host-side pass).
    float s = 0.0f;
    for (int i = lane; i < np; i += 32) s += partials[i];
    #pragma unroll
    for (int off = 16; off > 0; off >>= 1)
        s += __shfl_down(s, off, 32);
    if (lane == 0) out[0] = s;
#endif
}

extern "C" void kernel_launch(void* const* d_in, const int* in_sizes, int n_in,
                              void* d_out, int out_size, void* d_ws, size_t ws_size,
                              hipStream_t stream) {
    const float* pred = (const float*)d_in[0];
    const float* sdf  = (const float*)d_in[1];
    float* out        = (float*)d_out;
    float* partials   = (float*)d_ws;   // S1_BLOCKS floats = 8 KB

    pinn_stage1<<<S1_BLOCKS, S1_THREADS, 0, stream>>>(pred, sdf, partials);
    pinn_stage2<<<1, 32, 0, stream>>>(partials, out, S1_BLOCKS);
}